// YOLOv8Wrapper_7919919694133
// MI455X (gfx1250) — compile-verified
//
#include <hip/hip_runtime.h>
#include <stdint.h>

// Problem constants (from reference setup_inputs)
#define BB 64
#define CC 256
#define HH 80
#define WW 80
#define NN 100
#define IMGSZ 640.0f

// LDS padding: +1 dword every 16 dwords (done by TDM pad_enable on load).
// Rows are 80 dwords = 5 pad intervals -> padded row stride = 85 dwords.
// 85 mod 64 = 21, coprime with 64 -> row-wise pass is bank-conflict free;
// x + (x>>4) is injective mod 64 over a wave -> column pass conflict-free too.
#define PROW 85
__device__ __forceinline__ int pidx(int y, int x) { return y * PROW + x + (x >> 4); }

typedef unsigned int v4u __attribute__((ext_vector_type(4)));
typedef int          v4i __attribute__((ext_vector_type(4)));
typedef int          v8i __attribute__((ext_vector_type(8)));

// One workgroup (128 threads) per (b, c) feature plane.
//  0) Hoisted: box load + coordinate math (overlaps the DMA)
//  1) TDM tensor load: 80x80 f32 plane HBM -> LDS with hardware padding
//  2) In-LDS 2D inclusive prefix sum (cumsum over y, then x — same order as ref)
//  3) 100 boxes: 4 LDS lookups + inclusion/exclusion, mean, scatter-store
__global__ __launch_bounds__(128) void roi_mean_pool_kernel(
    const float* __restrict__ fmap,
    const float* __restrict__ boxes,
    float* __restrict__ out)
{
    __shared__ float P[HH * PROW];   // padded plane, then in-place 2D prefix

    const int tid   = threadIdx.x;
    const int plane = blockIdx.x;          // b*CC + c
    const int b     = plane >> 8;          // CC == 256
    const int c     = plane & (CC - 1);
    const float* src = fmap + (size_t)plane * (HH * WW);

    // ---- Hoisted per-box coordinate math: overlaps TDM DMA + prefix passes.
    int x1 = 0, y1 = 0, x2 = 0, y2 = 0;
    float inv_area = 0.0f;
    bool valid = false;
    if (tid < NN) {
        const float4 bx = reinterpret_cast<const float4*>(boxes)[b * NN + tid];
        // Match reference exactly: (coord / 640) * dim, trunc toward zero, clip.
        x1 = (int)(bx.x / IMGSZ * (float)WW);
        y1 = (int)(bx.y / IMGSZ * (float)HH);
        x2 = (int)(bx.z / IMGSZ * (float)WW);
        y2 = (int)(bx.w / IMGSZ * (float)HH);
        x1 = min(max(x1, 0), WW);  x2 = min(max(x2, 0), WW);
        y1 = min(max(y1, 0), HH);  y2 = min(max(y2, 0), HH);
        const int dy = y2 - y1;
        const int dx = x2 - x1;
        valid = (dy > 0) && (dx > 0);
        inv_area = 1.0f / (float)max(dy * dx, 1);
    }

#if __has_builtin(__builtin_amdgcn_tensor_load_to_lds)
    // ---- Tensor Data Mover path (CDNA5): one DMA per plane, issued by wave 0
    if (tid < 32) {
        const uint32_t lds_off = (uint32_t)(uintptr_t)(&P[0]); // low 32b = LDS byte offset
        const uint64_t ga      = (uint64_t)(uintptr_t)src;

        // D# group 0 (128b): count=1 | lds_addr | global_addr[56:0] | type=2
        v4u g0;
        g0[0] = 1u;                                            // count=1, user mode
        g0[1] = lds_off;                                       // lds_addr
        g0[2] = (uint32_t)ga;                                  // global_addr[31:0]
        g0[3] = (uint32_t)((ga >> 32) & 0x01FFFFFFu)           // global_addr[56:32]
              | (2u << 30);                                    // type = 2 ("image")

        // D# group 1 (256b): data_size=4B, pad 1 dword per 16 dwords,
        // tensor 80x80, tile 80x80, stride0=80
        v8i g1;
        g1[0] = (int)((2u << 16)                 // data_size code 2 (4 bytes)
                    | (1u << 20)                 // pad_enable
                    | (3u << 22)                 // pad_interval code 3 = 16 dwords
                    | (0u << 25));               // pad_amount code 0 = 1 dword
        g1[1] = (int)((uint32_t)WW << 16);       // tensor_dim0[15:0] @ bits 63:48
        g1[2] = (int)((uint32_t)HH << 16);       // tensor_dim0[31:16]=0 | tensor_dim1[15:0]
        g1[3] = (int)((uint32_t)WW << 16);       // tensor_dim1[31:16]=0 | tile_dim0=80
        g1[4] = (int)HH;                         // tile_dim1=80 | tile_dim2=0
        g1[5] = (int)WW;                         // tensor_dim0_stride[31:0]=80
        g1[6] = 0;                               // stride0 hi | dim1_stride lo (unused, 2D)
        g1[7] = 0;

        v4i gz4; gz4[0] = 0; gz4[1] = 0; gz4[2] = 0; gz4[3] = 0;   // groups 2/3 unused (2D)
        v8i gz8; gz8[0] = 0; gz8[1] = 0; gz8[2] = 0; gz8[3] = 0;
                 gz8[4] = 0; gz8[5] = 0; gz8[6] = 0; gz8[7] = 0;

        __builtin_amdgcn_tensor_load_to_lds(g0, g1, gz4, gz4, gz8, 0 /*cpol*/);
        __builtin_amdgcn_s_wait_tensorcnt(0);
    }
#else
    // Fallback: cooperative loads into the same padded layout
    for (int i = tid; i < HH * WW; i += 128) P[i + (i >> 4)] = src[i];
#endif
    __syncthreads();

    // ---- Pass 1: inclusive cumsum along y (reference axis=2), thread t = column t.
    if (tid < WW) {
        int a = tid + (tid >> 4);
        float acc = 0.0f;
        for (int y = 0; y < HH; ++y, a += PROW) {
            acc += P[a];
            P[a] = acc;
        }
    }
    __syncthreads();

    // ---- Pass 2: inclusive cumsum along x (reference axis=3), thread t = row t.
    if (tid < HH) {
        float* row = &P[tid * PROW];
        float acc = 0.0f;
        for (int x = 0; x < WW; ++x) {
            const int a = x + (x >> 4);
            acc += row[a];
            row[a] = acc;
        }
    }
    __syncthreads();

    // ---- Per-box lookup: P is the inclusive 2D prefix; virtual zero border at idx -1.
    if (tid < NN) {
        auto pref = [&](int y, int x) -> float {
            return (y < 0 || x < 0) ? 0.0f : P[pidx(y, x)];
        };

        const float s = pref(y2 - 1, x2 - 1) - pref(y1 - 1, x2 - 1)
                      - pref(y2 - 1, x1 - 1) + pref(y1 - 1, x1 - 1);

        out[((size_t)(b * NN + tid)) * CC + c] = valid ? (s * inv_area) : 0.0f;
    }
}

extern "C" void kernel_launch(void* const* d_in, const int* in_sizes, int n_in,
                              void* d_out, int out_size, void* d_ws, size_t ws_size,
                              hipStream_t stream) {
    (void)in_sizes; (void)n_in; (void)d_ws; (void)ws_size; (void)out_size;
    const float* fmap  = (const float*)d_in[0];   // [64,256,80,80] f32
    const float* boxes = (const float*)d_in[1];   // [64,100,4]     f32
    float*       out   = (float*)d_out;           // [64,100,256]   f32

    roi_mean_pool_kernel<<<BB * CC, 128, 0, stream>>>(fmap, boxes, out);
}